// IDWT2D_6433861009800
// MI455X (gfx1250) — compile-verified
//
#include <hip/hip_runtime.h>

// ---------------------------------------------------------------------------
// Inverse 2D Haar DWT (pywt 'haar'): out 2x2 block = (+-a +-b +-c +-d)/2.
// Memory-bound (AI ~0.37 flop/byte): target is the 23.3 TB/s HBM stream
// (~537 MB total traffic -> ~23 us floor). CDNA5 path: double-buffered
// global->LDS async copies (ASYNCcnt) feeding a float4 butterfly; outputs
// leave via non-temporal b128 stores (write-once stream >> 192 MB L2).
// ---------------------------------------------------------------------------

typedef float v4f __attribute__((ext_vector_type(4)));

#define H_IN      128
#define W_IN      128
#define W_OUT     256
#define TILE_ROWS 8
#define NBLK      (H_IN / TILE_ROWS)          // 16 row-blocks per plane
#define SUB_ELEMS (TILE_ROWS * W_IN)          // 1024 floats per subband tile
#define BUF_ELEMS (4 * SUB_ELEMS)             // 4 subbands -> 16 KB per buffer

// ---- async global -> LDS b128 copy (CDNA5: GLOBAL_LOAD_ASYNC_TO_LDS_B128) --
// The "vdst" operand is the wave-relative LDS byte offset; a flat pointer into
// LDS has addr[31:0] == that offset (ISA aperture rule), so truncation is exact.
__device__ __forceinline__ void async_b128(const float* g, float* l) {
  unsigned lds_off = (unsigned)(unsigned long long)l;   // addr[31:0] = LDS offset
  asm volatile("global_load_async_to_lds_b128 %0, %1, off"
               :: "v"(lds_off), "v"(g) : "memory");
}

// ---- s_wait_asynccnt ------------------------------------------------------
#if defined(__gfx1250__) && __has_builtin(__builtin_amdgcn_s_wait_asynccnt)
#define ASYNC_WAIT(n) __builtin_amdgcn_s_wait_asynccnt(n)
#else
#define ASYNC_WAIT(n) asm volatile("s_wait_asynccnt %0" :: "n"(n) : "memory")
#endif

__global__ void __launch_bounds__(256)
idwt2_haar_kernel(const float* __restrict__ LL, const float* __restrict__ LH,
                  const float* __restrict__ HL, const float* __restrict__ HH,
                  float* __restrict__ out) {
  __shared__ float smem[2][BUF_ELEMS];        // 32 KB of 320 KB/WGP

  const int t  = threadIdx.x;                 // 256 threads = 8 wave32
  const int tr = t >> 5;                      // tile row 0..7 (uniform per wave)
  const int tc = t & 31;                      // col group 0..31 (lane id)
  const int bc = blockIdx.x;                  // plane index (B*C)

  const long inPlane  = (long)bc * (H_IN * W_IN);
  const long outPlane = (long)bc * (2 * H_IN) * W_OUT;

  const int lsub = tr * W_IN + 4 * tc;        // float offset inside subband tile

  const float* __restrict__ subs[4] = {LL, LH, HL, HH};

  auto issue = [&](int buf, int blk) {
    const long g = inPlane + (long)(blk * TILE_ROWS + tr) * W_IN + 4 * tc;
    float* base = &smem[buf][0];
    async_b128(subs[0] + g, base + 0 * SUB_ELEMS + lsub);
    async_b128(subs[1] + g, base + 1 * SUB_ELEMS + lsub);
    async_b128(subs[2] + g, base + 2 * SUB_ELEMS + lsub);
    async_b128(subs[3] + g, base + 3 * SUB_ELEMS + lsub);
  };

  issue(0, 0);                                // prime the pipeline

  for (int blk = 0; blk < NBLK; ++blk) {
    const int buf = blk & 1;
    if (blk + 1 < NBLK) {
      issue(buf ^ 1, blk + 1);                // prefetch next tile (other buffer)
      ASYNC_WAIT(4);                          // 8 outstanding; first 4 (= this
                                              // block) done: async loads retire
                                              // in order
    } else {
      ASYNC_WAIT(0);                          // drain the final tile
    }
    __syncthreads();                          // all waves' copies landed

    const float* base = &smem[buf][0];
    v4f a = *(const v4f*)(base + 0 * SUB_ELEMS + lsub);   // ds_load_b128
    v4f b = *(const v4f*)(base + 1 * SUB_ELEMS + lsub);
    v4f c = *(const v4f*)(base + 2 * SUB_ELEMS + lsub);
    v4f d = *(const v4f*)(base + 3 * SUB_ELEMS + lsub);

    // 4-point Haar butterfly, 1/sqrt2 applied twice -> 0.5
    v4f p = a + b, m = a - b, q = c + d, n = c - d;
    v4f e0 = (p + q) * 0.5f;                  // out[2r,   2k  ]
    v4f o0 = (m + n) * 0.5f;                  // out[2r,   2k+1]
    v4f e1 = (p - q) * 0.5f;                  // out[2r+1, 2k  ]
    v4f o1 = (m - n) * 0.5f;                  // out[2r+1, 2k+1]

    const int orow = 2 * (blk * TILE_ROWS + tr);
    float* r0 = out + outPlane + (long)orow * W_OUT + 8 * tc;
    float* r1 = r0 + W_OUT;

    v4f s00 = {e0.x, o0.x, e0.y, o0.y};
    v4f s01 = {e0.z, o0.z, e0.w, o0.w};
    v4f s10 = {e1.x, o1.x, e1.y, o1.y};
    v4f s11 = {e1.z, o1.z, e1.w, o1.w};
    // Write-once output stream (256 MB > L2): non-temporal b128 stores.
    __builtin_nontemporal_store(s00, (v4f*)(r0 + 0));
    __builtin_nontemporal_store(s01, (v4f*)(r0 + 4));
    __builtin_nontemporal_store(s10, (v4f*)(r1 + 0));
    __builtin_nontemporal_store(s11, (v4f*)(r1 + 4));

    __syncthreads();                          // buffer reusable for blk+2 copies
  }
}

extern "C" void kernel_launch(void* const* d_in, const int* in_sizes, int n_in,
                              void* d_out, int out_size, void* d_ws, size_t ws_size,
                              hipStream_t stream) {
  const float* LL = (const float*)d_in[0];
  const float* LH = (const float*)d_in[1];
  const float* HL = (const float*)d_in[2];
  const float* HH = (const float*)d_in[3];
  float* out = (float*)d_out;

  const int planes = in_sizes[0] / (H_IN * W_IN);   // B*C = 1024
  hipLaunchKernelGGL(idwt2_haar_kernel, dim3(planes), dim3(256), 0, stream,
                     LL, LH, HL, HH, out);
}